// LILinearWeights_27075473834355
// MI455X (gfx1250) — compile-verified
//
#include <hip/hip_runtime.h>
#include <hip/hip_bf16.h>

// ---------------------------------------------------------------------------
// Problem constants (fixed by setup_inputs): S=32, O=1024, P=512, fp32.
// prec / L / Linv / mu are sample-independent (Xi broadcast over S), so we
// factor once and turn every triangular solve into a WMMA fp32 GEMM.
// ---------------------------------------------------------------------------
#define SDIM 32
#define ODIM 1024
#define PDIM 512

// GEMM tiling: 128x128 block tile, K-chunk 32, 256 threads = 8 wave32 waves.
// Wave grid 4(m) x 2(n); each wave owns 2x4 subtiles of 16x16 (v8f accums).
#define BM 128
#define BN 128
#define BK 32
#define KSTR (BK + 4)          // 36 floats per LDS row (k contiguous, 16B aligned)
#define NTHREADS 256
#define TILE_F (BM * KSTR)     // 4608 floats per staged operand

typedef __attribute__((ext_vector_type(2))) float v2f;
typedef __attribute__((ext_vector_type(8))) float v8f;

__device__ __forceinline__ v8f wmma_f32(v2f a, v2f b, v8f c) {
  // D = A(16x4) * B(4x16) + C(16x16), all fp32.
  return __builtin_amdgcn_wmma_f32_16x16x4_f32(false, a, false, b,
                                               (short)0, c, false, false);
}

// ---- CDNA5 async global->LDS copy (ASYNCcnt path) -------------------------
// LDS byte address = low 32 bits of generic shared pointer (flat aperture).
__device__ __forceinline__ void async_ld16(const float* g, const float* l) {
  asm volatile("global_load_async_to_lds_b128 %0, %1, off"
               :
               : "v"((unsigned)(unsigned long long)l),
                 "v"((unsigned long long)g)
               : "memory");
}
__device__ __forceinline__ void wait_async0() {
#if __has_builtin(__builtin_amdgcn_s_wait_asynccnt)
  __builtin_amdgcn_s_wait_asynccnt(0);
#else
  asm volatile("s_wait_asynccnt 0" ::: "memory");
#endif
}

// ---- LDS staging ----------------------------------------------------------
// Both operands land k-contiguous: S[row][k], row stride KSTR. A-frag and
// B-frag are then both single contiguous b64 loads per lane.

// 128 rows x 32 k, source row-major with row stride ldg -> pure 16B copies.
__device__ __forceinline__ void stage_rows_async(const float* __restrict__ G,
                                                 int ldg, int r0, int kk,
                                                 float* __restrict__ S) {
  const int tid = threadIdx.x;
#pragma unroll
  for (int p = 0; p < (BM * BK) / (4 * NTHREADS); ++p) {
    const int idx = tid + p * NTHREADS;
    const int rr = idx >> 3;             // 8 float4 per row
    const int k4 = (idx & 7) << 2;
    async_ld16(G + (size_t)(r0 + rr) * ldg + kk + k4, S + rr * KSTR + k4);
  }
}

// Source stored K x R row-major (transpose case): scatter scalar writes.
__device__ __forceinline__ void stage_rows_scatter(const float* __restrict__ G,
                                                   int ldg, int r0, int kk,
                                                   float* __restrict__ S) {
  const int tid = threadIdx.x;
#pragma unroll
  for (int p = 0; p < (BM * BK) / (4 * NTHREADS); ++p) {
    const int idx = tid + p * NTHREADS;
    const int kr = idx >> 5;             // 32 float4 per k-row
    const int r4 = (idx & 31) << 2;
    const float4 v = *(const float4*)(G + (size_t)(kk + kr) * ldg + r0 + r4);
    S[(r4 + 0) * KSTR + kr] = v.x;
    S[(r4 + 1) * KSTR + kr] = v.y;
    S[(r4 + 2) * KSTR + kr] = v.z;
    S[(r4 + 3) * KSTR + kr] = v.w;
  }
}

// One BK chunk of WMMA MACs for this wave's 2x4 subtiles.
__device__ __forceinline__ void mma_chunk(const float* __restrict__ As,
                                          const float* __restrict__ Bs,
                                          int wm0, int wn0, int lr, int half,
                                          v8f acc[2][4]) {
#pragma unroll
  for (int k4 = 0; k4 < BK / 4; ++k4) {
    const int k0 = k4 * 4;
    v2f a[2];
#pragma unroll
    for (int sm = 0; sm < 2; ++sm)
      // A 16x4 layout: lanes<16 hold K=k0,k0+1; lanes>=16 hold K=k0+2,k0+3.
      a[sm] = *(const v2f*)(As + (wm0 + sm * 16 + lr) * KSTR + k0 + 2 * half);
    v2f b[4];
#pragma unroll
    for (int sn = 0; sn < 4; ++sn)
      b[sn] = *(const v2f*)(Bs + (wn0 + sn * 16 + lr) * KSTR + k0 + 2 * half);
#pragma unroll
    for (int sm = 0; sm < 2; ++sm)
#pragma unroll
      for (int sn = 0; sn < 4; ++sn)
        acc[sm][sn] = wmma_f32(a[sm], b[sn], acc[sm][sn]);
  }
}

// ---- Generic WMMA GEMM: C[M,N] = opA(A) @ opB(B) (+ diag) -----------------
// AT: A stored K x M (else M x K). BT: B stored N x K (else K x N).
template <bool AT, bool BT, bool DIAG>
__global__ __launch_bounds__(NTHREADS) void gemm_kernel(
    const float* __restrict__ A, const float* __restrict__ B,
    float* __restrict__ C, int K, int lda, int ldb, int ldc, float diagval) {
  __shared__ float As[TILE_F];
  __shared__ float Bs[TILE_F];
  const int m0 = blockIdx.x * BM;
  const int n0 = blockIdx.y * BN;
  const int tid = threadIdx.x;
  const int wid = tid >> 5, lane = tid & 31;
  const int lr = lane & 15, half = lane >> 4;
  const int wm0 = (wid & 3) * 32;
  const int wn0 = (wid >> 2) * 64;

  v8f acc[2][4] = {};
  for (int kk = 0; kk < K; kk += BK) {
    if constexpr (!AT) stage_rows_async(A, lda, m0, kk, As);
    else               stage_rows_scatter(A, lda, m0, kk, As);
    if constexpr (BT)  stage_rows_async(B, ldb, n0, kk, Bs);
    else               stage_rows_scatter(B, ldb, n0, kk, Bs);
    if constexpr (AT && !BT) {  // prefetch next chunk for the sync paths
      if (kk + BK < K) {
        __builtin_prefetch(A + (size_t)(kk + BK) * lda + m0 + (size_t)tid * 16, 0, 1);
        __builtin_prefetch(B + (size_t)(kk + BK) * ldb + n0 + (size_t)tid * 16, 0, 1);
      }
    }
    if constexpr (!AT || BT) wait_async0();
    __syncthreads();
    mma_chunk(As, Bs, wm0, wn0, lr, half, acc);
    __syncthreads();
  }
  // Epilogue: C/D layout — VGPR r: lanes<16 -> M=r, lanes>=16 -> M=r+8.
#pragma unroll
  for (int sm = 0; sm < 2; ++sm)
#pragma unroll
    for (int sn = 0; sn < 4; ++sn) {
      const int gn = n0 + wn0 + sn * 16 + lr;
#pragma unroll
      for (int r = 0; r < 8; ++r) {
        const int gm = m0 + wm0 + sm * 16 + r + 8 * half;
        float v = acc[sm][sn][r];
        if (DIAG && gm == gn) v += diagval;
        C[(size_t)gm * ldc + gn] = v;
      }
    }
}

// ---- Sampler: sample[s] = (mu + LinvT @ Z_s^T)^T, plus reductions ---------
// Double-buffered: next K-chunk streams in via async-to-LDS while WMMA
// consumes the current one.
__global__ __launch_bounds__(NTHREADS) void sample_kernel(
    const float* __restrict__ LinvT, const float* __restrict__ Z,
    const float* __restrict__ mu, float* __restrict__ out,
    float* __restrict__ acc) {
  __shared__ float lds[4 * TILE_F];  // {As0,Bs0,As1,Bs1}; Ct overlays lds[0..]
  __shared__ float red[NTHREADS];

  const int s  = blockIdx.z;
  const int m0 = blockIdx.x * BM;      // p-tile
  const int n0 = blockIdx.y * BN;      // o-tile
  const int tid = threadIdx.x;
  const int wid = tid >> 5, lane = tid & 31;
  const int lr = lane & 15, half = lane >> 4;
  const int wm0 = (wid & 3) * 32;
  const int wn0 = (wid >> 2) * 64;

  const float* Zs = Z + (size_t)s * ODIM * PDIM;  // stored [O][P] = N x K
  const bool do_zss = (blockIdx.x == 0);          // count each Z element once

  // Prologue: stage chunk 0 into buffer 0.
  stage_rows_async(LinvT, PDIM, m0, 0, lds);
  stage_rows_async(Zs, PDIM, n0, 0, lds + TILE_F);
  wait_async0();
  __syncthreads();

  float zss = 0.0f;
  v8f acc8[2][4] = {};
  int cur = 0;
  for (int kk = 0; kk < PDIM; kk += BK) {
    float* Acur = lds + (2 * cur) * TILE_F;
    float* Bcur = lds + (2 * cur + 1) * TILE_F;
    if (kk + BK < PDIM) {  // stream next chunk into the other buffer
      float* Anxt = lds + (2 * (cur ^ 1)) * TILE_F;
      float* Bnxt = lds + (2 * (cur ^ 1) + 1) * TILE_F;
      stage_rows_async(LinvT, PDIM, m0, kk + BK, Anxt);
      stage_rows_async(Zs, PDIM, n0, kk + BK, Bnxt);
    }
    mma_chunk(Acur, Bcur, wm0, wn0, lr, half, acc8);
    if (do_zss) {  // sum(Z^2) from the staged tile (uniform within block)
#pragma unroll
      for (int p = 0; p < (BN * BK) / NTHREADS; ++p) {
        const int idx = tid + p * NTHREADS;
        const float v = Bcur[(idx >> 5) * KSTR + (idx & 31)];
        zss += v * v;
      }
    }
    wait_async0();
    __syncthreads();
    cur ^= 1;
  }

  // Epilogue: add mu, transpose through LDS in two n-halves (64x128 each),
  // store coalesced to out[s][o][p], accumulate sum(sample^2).
  float* Ct = lds;  // 64 x (BM+4) = 8448 floats
  float* outS = out + (size_t)s * ODIM * PDIM;
  float ssq = 0.0f;
#pragma unroll
  for (int h = 0; h < 2; ++h) {
    if ((wid >> 2) == h) {
#pragma unroll
      for (int sm = 0; sm < 2; ++sm)
#pragma unroll
        for (int sn = 0; sn < 4; ++sn) {
          const int nl = sn * 16 + lr;                  // local n within half
#pragma unroll
          for (int r = 0; r < 8; ++r) {
            const int ml = wm0 + sm * 16 + r + 8 * half;  // local m (p)
            const float v = acc8[sm][sn][r] +
                mu[(size_t)(m0 + ml) * ODIM + (n0 + 64 * h + nl)];
            ssq += v * v;
            Ct[nl * (BM + 4) + ml] = v;
          }
        }
    }
    __syncthreads();
#pragma unroll
    for (int p = 0; p < (64 * BM) / (4 * NTHREADS); ++p) {  // 8 float4/thread
      const int idx = tid + p * NTHREADS;
      const int nl = idx >> 5;
      const int m4 = (idx & 31) << 2;
      const float4 v = *(const float4*)(Ct + nl * (BM + 4) + m4);
      *(float4*)(outS + (size_t)(n0 + 64 * h + nl) * PDIM + m0 + m4) = v;
    }
    __syncthreads();
  }

  // Block reductions -> global atomics.
  red[tid] = ssq;
  __syncthreads();
  for (int off = NTHREADS / 2; off > 0; off >>= 1) {
    if (tid < off) red[tid] += red[tid + off];
    __syncthreads();
  }
  if (tid == 0) atomicAdd(&acc[s], red[0]);
  __syncthreads();
  red[tid] = zss;
  __syncthreads();
  for (int off = NTHREADS / 2; off > 0; off >>= 1) {
    if (tid < off) red[tid] += red[tid + off];
    __syncthreads();
  }
  if (tid == 0 && do_zss) atomicAdd(&acc[SDIM + s], red[0]);
}

// ---- Xs = exp(log_prec) * Xi (row-scaled), also zero accumulators ---------
__global__ __launch_bounds__(NTHREADS) void scale_kernel(
    const float* __restrict__ Xi, const float* __restrict__ lps,
    float* __restrict__ Xs, float* __restrict__ acc) {
  const int idx = blockIdx.x * NTHREADS + threadIdx.x;
  const int q = idx >> 9;  // row index (P=512)
  const float lam = __expf(lps[q]);  // LOG_PREC_LR == 1.0
  Xs[idx] = lam * Xi[idx];
  if (blockIdx.x == 0 && threadIdx.x < 2 * SDIM) acc[threadIdx.x] = 0.0f;
}

// ---- Blocked right-looking Cholesky, single workgroup, panel in LDS -------
__global__ __launch_bounds__(NTHREADS) void cholesky_kernel(float* __restrict__ A) {
  const int tid = threadIdx.x;
  __shared__ float panel[512 * 16];  // 32 KB
  __shared__ float dinv_s;
  const int NB = 16;
  for (int j = 0; j < PDIM; j += NB) {
    const int rows = PDIM - j;
    for (int idx = tid; idx < rows * NB; idx += NTHREADS) {
      const int r = idx >> 4, c = idx & 15;
      panel[idx] = A[(size_t)(j + r) * PDIM + j + c];
    }
    __syncthreads();
    for (int c = 0; c < NB; ++c) {
      if (tid == 0) {
        const float d = sqrtf(panel[c * NB + c]);
        panel[c * NB + c] = d;
        dinv_s = 1.0f / d;
      }
      __syncthreads();
      const float dinv = dinv_s;
      for (int r = c + 1 + tid; r < rows; r += NTHREADS) panel[r * NB + c] *= dinv;
      __syncthreads();
      for (int k = c + 1; k < NB; ++k) {
        const float lkc = panel[k * NB + c];
        for (int r = k + tid; r < rows; r += NTHREADS)
          panel[r * NB + k] -= panel[r * NB + c] * lkc;
      }
      __syncthreads();
    }
    for (int idx = tid; idx < rows * NB; idx += NTHREADS) {
      const int r = idx >> 4, c = idx & 15;
      A[(size_t)(j + r) * PDIM + j + c] = panel[idx];
    }
    __syncthreads();
    const int t0 = j + NB;
    for (int i = t0 + tid; i < PDIM; i += NTHREADS) {
      const float* pr = &panel[(i - j) * NB];
      float* arow = &A[(size_t)i * PDIM];
      for (int k = t0; k <= i; ++k) {
        const float* pk = &panel[(k - j) * NB];
        float ssum = 0.0f;
#pragma unroll
        for (int c = 0; c < NB; ++c) ssum += pr[c] * pk[c];
        arow[k] -= ssum;
      }
    }
    __syncthreads();
  }
}

// ---- Linv = L^-1 (forward substitution, one column per thread) ------------
__global__ __launch_bounds__(NTHREADS) void trinv_kernel(
    const float* __restrict__ L, float* __restrict__ Linv,
    float* __restrict__ LinvT) {
  const int c = blockIdx.x * NTHREADS + threadIdx.x;
  float* xrow = LinvT + (size_t)c * PDIM;  // LinvT[c][i] = Linv[i][c]
  for (int i = 0; i < c; ++i) {
    Linv[(size_t)i * PDIM + c] = 0.0f;
    xrow[i] = 0.0f;
  }
  const float xc = 1.0f / L[(size_t)c * PDIM + c];
  Linv[(size_t)c * PDIM + c] = xc;
  xrow[c] = xc;
  for (int i = c + 1; i < PDIM; ++i) {
    const float* Lrow = L + (size_t)i * PDIM;
    float s = 0.0f;
    for (int k = c; k < i; ++k) s += Lrow[k] * xrow[k];
    const float xi = -s / Lrow[i];
    Linv[(size_t)i * PDIM + c] = xi;
    xrow[i] = xi;
  }
}

// ---- logpq[s] = logP - logQ ----------------------------------------------
__global__ __launch_bounds__(NTHREADS) void finalize_kernel(
    const float* __restrict__ L, const float* __restrict__ acc,
    float* __restrict__ logpq) {
  __shared__ float red[NTHREADS];
  const int tid = threadIdx.x;
  float s = 0.0f;
  for (int j = tid; j < PDIM; j += NTHREADS) s += logf(L[(size_t)j * PDIM + j]);
  red[tid] = s;
  __syncthreads();
  for (int off = NTHREADS / 2; off > 0; off >>= 1) {
    if (tid < off) red[tid] += red[tid + off];
    __syncthreads();
  }
  const float logdet_single = 2.0f * red[0];
  if (tid < SDIM) {
    const float sumsq = acc[tid];
    const float zsq = acc[SDIM + tid];
    const float Pf = (float)PDIM, Of = (float)ODIM;
    const float logP = -0.5f * Pf * sumsq + 0.5f * Of * Pf * logf(Pf);
    const float logQ = -0.5f * zsq + 0.5f * Of * logdet_single;
    logpq[tid] = logP - logQ;
  }
}

// ---------------------------------------------------------------------------
extern "C" void kernel_launch(void* const* d_in, const int* in_sizes, int n_in,
                              void* d_out, int out_size, void* d_ws,
                              size_t ws_size, hipStream_t stream) {
  (void)in_sizes; (void)n_in; (void)out_size; (void)ws_size;
  // inputs: Xi_in[0] (unused), Xi_param[1] P*P, u[2] O*P, log_prec_scaled[3] P, Z[4] S*O*P
  const float* Xi  = (const float*)d_in[1];
  const float* u   = (const float*)d_in[2];
  const float* lps = (const float*)d_in[3];
  const float* Z   = (const float*)d_in[4];
  float* out = (float*)d_out;  // sample [S,O,P] then logpq [S]
  float* ws = (float*)d_ws;    // needs ~10.5 MB

  float* Xs    = ws;                 // P*P
  float* Lm    = ws + 262144;        // P*P (prec -> L in place)
  float* Linv  = ws + 524288;        // P*P
  float* LinvT = ws + 786432;        // P*P
  float* Y     = ws + 1048576;       // P*O
  float* T1    = ws + 1572864;       // P*O
  float* mu    = ws + 2097152;       // P*O
  float* acc   = ws + 2621440;       // 2*S floats

  // Xs = diag(exp(log_prec)) * Xi ; zero accumulators
  scale_kernel<<<(PDIM * PDIM) / NTHREADS, NTHREADS, 0, stream>>>(Xi, lps, Xs, acc);
  // prec = Xs^T @ Xi + P*I
  gemm_kernel<true, false, true><<<dim3(PDIM / BM, PDIM / BN), NTHREADS, 0, stream>>>(
      Xs, Xi, Lm, PDIM, PDIM, PDIM, PDIM, (float)PDIM);
  // Y = Xs^T @ u^T   (u stored O x P)
  gemm_kernel<true, true, false><<<dim3(PDIM / BM, ODIM / BN), NTHREADS, 0, stream>>>(
      Xs, u, Y, PDIM, PDIM, PDIM, ODIM, 0.0f);
  // L = chol(prec), in place
  cholesky_kernel<<<1, NTHREADS, 0, stream>>>(Lm);
  // Linv = L^-1 (and its transpose)
  trinv_kernel<<<PDIM / NTHREADS, NTHREADS, 0, stream>>>(Lm, Linv, LinvT);
  // T1 = Linv @ Y ; mu = Linv^T @ T1
  gemm_kernel<false, false, false><<<dim3(PDIM / BM, ODIM / BN), NTHREADS, 0, stream>>>(
      Linv, Y, T1, PDIM, PDIM, ODIM, ODIM, 0.0f);
  gemm_kernel<false, false, false><<<dim3(PDIM / BM, ODIM / BN), NTHREADS, 0, stream>>>(
      LinvT, T1, mu, PDIM, PDIM, ODIM, ODIM, 0.0f);
  // sample[s] = (mu + LinvT @ Z_s^T)^T, plus sum(sample^2), sum(Z^2)
  sample_kernel<<<dim3(PDIM / BM, ODIM / BN, SDIM), NTHREADS, 0, stream>>>(
      LinvT, Z, mu, out, acc);
  // logpq
  finalize_kernel<<<1, NTHREADS, 0, stream>>>(Lm, acc, out + (size_t)SDIM * ODIM * PDIM);
}